// E3Conv_50070728736822
// MI455X (gfx1250) — compile-verified
//
#include <hip/hip_runtime.h>
#include <hip/hip_bf16.h>
#include <math.h>

#define N_NODES 20000
#define N_EDGES 320000

typedef float v2f __attribute__((ext_vector_type(2)));
typedef float v8f __attribute__((ext_vector_type(8)));

__device__ __forceinline__ v8f wmma_f32(v2f a, v2f b, v8f c) {
    // V_WMMA_F32_16X16X4_F32 : D(16x16,f32) = A(16x4,f32) x B(4x16,f32) + C
    return __builtin_amdgcn_wmma_f32_16x16x4_f32(
        /*neg_a=*/false, a, /*neg_b=*/false, b,
        /*c_mod=*/(short)0, c, /*reuse_a=*/false, /*reuse_b=*/false);
}

__device__ __forceinline__ float silu(float x) {
    return x / (1.0f + __expf(-x));
}

// Packed-B layout: pk[((ks*2 + hi)*N + n)*2 + j] = W[(ks*4 + hi*2 + j)*N + n]
// -> per-lane B fragment (rows k0,k0+1 at col nn) is one aligned b64 load.
__device__ __forceinline__ v2f loadB(const float* __restrict__ pk, int ks, int hi,
                                     int nn, int Ncols) {
    return *(const v2f*)&pk[(((ks * 2 + hi) * Ncols) + nn) * 2];
}

// ---------------------------------------------------------------------------
// Kernel 0: zero the accumulator region (ns | nv | deg are contiguous in ws)
// ---------------------------------------------------------------------------
__global__ __launch_bounds__(256) void e3_zero_kernel(float* __restrict__ p, int n) {
    int i = blockIdx.x * blockDim.x + threadIdx.x;
    if (i < n) p[i] = 0.0f;
}

// ---------------------------------------------------------------------------
// Kernel 0b: repack all weight matrices into WMMA B-fragment order.
// Segments (floats): w1_s 4096 | wsc_s 4096 | w1_v 4096 | wsc_v 4096 |
//                    w2_s 8192 | w2_v 8192 | fc1 512 | fc2 4096 | fc3 16384
// ---------------------------------------------------------------------------
__device__ __forceinline__ void pack_frag(const float* __restrict__ src,
                                          float* __restrict__ dst, int idx, int Ncols) {
    const int k = idx / Ncols;          // Ncols is compile-time const at call sites
    const int n = idx - k * Ncols;
    const int ks = k >> 2, rem = k & 3;
    const int hi2 = rem >> 1, j = rem & 1;
    dst[(((ks * 2 + hi2) * Ncols) + n) * 2 + j] = src[idx];
}

__global__ __launch_bounds__(256) void e3_pack_kernel(
    const float* __restrict__ w1_s, const float* __restrict__ wsc_s,
    const float* __restrict__ w1_v, const float* __restrict__ wsc_v,
    const float* __restrict__ w2_s, const float* __restrict__ w2_v,
    const float* __restrict__ fc1,  const float* __restrict__ fc2,
    const float* __restrict__ fc3,  float* __restrict__ pk)
{
    const int i = blockIdx.x * blockDim.x + threadIdx.x;
    if      (i < 4096)  pack_frag(w1_s,  pk + 0,     i,         64);
    else if (i < 8192)  pack_frag(wsc_s, pk + 4096,  i - 4096,  64);
    else if (i < 12288) pack_frag(w1_v,  pk + 8192,  i - 8192,  64);
    else if (i < 16384) pack_frag(wsc_v, pk + 12288, i - 12288, 64);
    else if (i < 24576) pack_frag(w2_s,  pk + 16384, i - 16384, 64);
    else if (i < 32768) pack_frag(w2_v,  pk + 24576, i - 24576, 64);
    else if (i < 33280) pack_frag(fc1,   pk + 32768, i - 32768, 64);
    else if (i < 37376) pack_frag(fc2,   pk + 33280, i - 33280, 64);
    else if (i < 53760) pack_frag(fc3,   pk + 37376, i - 37376, 256);
}

// ---------------------------------------------------------------------------
// Kernel 1: node GEMMs.  Block = 16 nodes, 8 waves.
//   wave 0: ys = xs @ w1_s   wave 1: sc_s = xs @ wsc_s
//   wave 2-4: yv[d] = xv[:,:,d] @ w1_v   wave 5-7: sc_v[d] = xv[:,:,d] @ wsc_v
// LDS staging de-interleaves the vector part (col 64+d*64+u) so every wave
// reads A fragments with unit stride (branchless b64 loads).
// y / sc global row layout stays interleaved: [0:64]=scalar, [64+u*3+d]=vec.
// ---------------------------------------------------------------------------
__global__ __launch_bounds__(256) void e3_node_gemm_kernel(
    const float* __restrict__ nh, const float* __restrict__ pk,
    float* __restrict__ y, float* __restrict__ sc)
{
    __shared__ __align__(16) float xbuf[16 * 260];  // stride 260: conflict-free
    const int n0 = blockIdx.x * 16;
    const int tid = threadIdx.x;

    for (int idx = tid; idx < 16 * 64; idx += 256) {
        int r = idx >> 6, c = idx & 63;
        xbuf[r * 260 + c] = nh[(n0 + r) * 256 + c];
    }
    for (int idx = tid; idx < 16 * 192; idx += 256) {
        int r = (unsigned)idx / 192u;
        int rem = idx - r * 192;
        int d = rem >> 6, u = rem & 63;
        xbuf[r * 260 + 64 + d * 64 + u] = nh[(n0 + r) * 256 + 64 + u * 3 + d];
    }
    __syncthreads();

    const int lane = tid & 31, wave = tid >> 5;
    const int l15 = lane & 15, hi = lane >> 4;

    const float* Wp;       // packed B
    float* outBase;
    int d = 0;
    bool isVec;
    if (wave == 0)      { Wp = pk + 0;     outBase = y;  isVec = false; }
    else if (wave == 1) { Wp = pk + 4096;  outBase = sc; isVec = false; }
    else if (wave < 5)  { Wp = pk + 8192;  outBase = y;  isVec = true; d = wave - 2; }
    else                { Wp = pk + 12288; outBase = sc; isVec = true; d = wave - 5; }

    const int colA0     = isVec ? 64 + d * 64 : 0;
    const int outColB   = isVec ? 64 + d : 0;
    const int outStride = isVec ? 3 : 1;
    const float inv = 0.125f;  // 1/sqrt(MUL)

    v2f a[16];
    #pragma unroll
    for (int ks = 0; ks < 16; ++ks)
        a[ks] = *(const v2f*)&xbuf[l15 * 260 + colA0 + ks * 4 + hi * 2];

    #pragma unroll
    for (int nt = 0; nt < 4; ++nt) {
        v8f acc = {};
        const int nn = nt * 16 + l15;
        #pragma unroll
        for (int ks = 0; ks < 16; ++ks)
            acc = wmma_f32(a[ks], loadB(Wp, ks, hi, nn, 64), acc);
        #pragma unroll
        for (int i = 0; i < 8; ++i) {
            const int m = i + hi * 8;
            outBase[(n0 + m) * 256 + outColB + nn * outStride] = acc[i] * inv;
        }
    }
}

// ---------------------------------------------------------------------------
// Kernel 2: edge MLP (WMMA) + gather + messages + atomic scatter.
// Block = 64 threads (2 waves); each wave owns an independent 16-edge tile.
// ---------------------------------------------------------------------------
__global__ __launch_bounds__(64) void e3_edge_kernel(
    const float* __restrict__ ea,  const float* __restrict__ esh,
    const int*   __restrict__ eidx, const float* __restrict__ pk,
    const float* __restrict__ y,
    float* __restrict__ ns, float* __restrict__ nv, float* __restrict__ deg)
{
    __shared__ __align__(16) float hbuf[2][16 * 68];    // per-wave 16x64 hidden
    __shared__ __align__(16) float wbuf[2][16 * 260];   // per-wave 16x256 weights

    const int tid = threadIdx.x;
    const int lane = tid & 31, wave = tid >> 5;
    const int l15 = lane & 15, hi = lane >> 4;
    const int e0 = (blockIdx.x * 2 + wave) * 16;
    float* hb = hbuf[wave];
    float* wb = wbuf[wave];

    const float* p_fc1 = pk + 32768;
    const float* p_fc2 = pk + 33280;
    const float* p_fc3 = pk + 37376;

    const float s1 = 0.35355339059327373f;  // 1/sqrt(NUM_BASIS)
    const float s2 = 0.125f;                // 1/sqrt(64)

    // ---- Layer 1: (16x8) @ fc_w1(8x64), silu -> hbuf ----
    {
        v2f a[2];
        #pragma unroll
        for (int ks = 0; ks < 2; ++ks)
            a[ks] = *(const v2f*)&ea[(e0 + l15) * 8 + ks * 4 + hi * 2];
        #pragma unroll
        for (int nt = 0; nt < 4; ++nt) {
            v8f acc = {};
            const int nn = nt * 16 + l15;
            #pragma unroll
            for (int ks = 0; ks < 2; ++ks)
                acc = wmma_f32(a[ks], loadB(p_fc1, ks, hi, nn, 64), acc);
            #pragma unroll
            for (int i = 0; i < 8; ++i)
                hb[(i + hi * 8) * 68 + nn] = silu(acc[i] * s1);
        }
    }
    __syncthreads();

    // ---- Layer 2: (16x64) @ fc_w2(64x64), silu -> hbuf (after all reads) ----
    {
        v2f a[16];
        #pragma unroll
        for (int ks = 0; ks < 16; ++ks)
            a[ks] = *(const v2f*)&hb[l15 * 68 + ks * 4 + hi * 2];
        v8f h2[4];
        #pragma unroll
        for (int nt = 0; nt < 4; ++nt) {
            v8f acc = {};
            const int nn = nt * 16 + l15;
            #pragma unroll
            for (int ks = 0; ks < 16; ++ks)
                acc = wmma_f32(a[ks], loadB(p_fc2, ks, hi, nn, 64), acc);
            h2[nt] = acc;
        }
        __syncthreads();
        #pragma unroll
        for (int nt = 0; nt < 4; ++nt)
            #pragma unroll
            for (int i = 0; i < 8; ++i)
                hb[(i + hi * 8) * 68 + nt * 16 + l15] = silu(h2[nt][i] * s2);
    }
    __syncthreads();

    // ---- Layer 3: (16x64) @ fc_w3(64x256) -> wbuf (wA|wB|wC|wD) ----
    {
        v2f a[16];
        #pragma unroll
        for (int ks = 0; ks < 16; ++ks)
            a[ks] = *(const v2f*)&hb[l15 * 68 + ks * 4 + hi * 2];
        #pragma unroll
        for (int nt = 0; nt < 16; ++nt) {
            v8f acc = {};
            const int nn = nt * 16 + l15;
            #pragma unroll
            for (int ks = 0; ks < 16; ++ks)
                acc = wmma_f32(a[ks], loadB(p_fc3, ks, hi, nn, 256), acc);
            #pragma unroll
            for (int i = 0; i < 8; ++i)
                wb[(i + hi * 8) * 260 + nn] = acc[i] * s2;
        }
    }
    __syncthreads();

    // ---- Message + scatter phase: lane = channel, loop edges ----
    for (int m = 0; m < 16; ++m) {
        const int e = e0 + m;
        const int srcN = eidx[e];
        const int dstN = eidx[N_EDGES + e];
        const float sh0  = esh[e * 4 + 0];
        const float sh1a = esh[e * 4 + 1];
        const float sh1b = esh[e * 4 + 2];
        const float sh1c = esh[e * 4 + 3];
        if (lane == 0) atomicAdd(&deg[srcN], 1.0f);
        const float* yrow  = y  + (size_t)dstN * 256;
        float*       nsrow = ns + (size_t)srcN * 128;
        float*       nvrow = nv + (size_t)srcN * 384;
        #pragma unroll
        for (int half = 0; half < 2; ++half) {
            const int u = half * 32 + lane;
            const float wA = wb[m * 260 + u];
            const float wB = wb[m * 260 + 64 + u];
            const float wC = wb[m * 260 + 128 + u];
            const float wD = wb[m * 260 + 192 + u];
            const float es  = yrow[u];
            const float ev0 = yrow[64 + u * 3 + 0];
            const float ev1 = yrow[64 + u * 3 + 1];
            const float ev2 = yrow[64 + u * 3 + 2];
            const float m0a = wA * es * sh0;
            const float dot = ev0 * sh1a + ev1 * sh1b + ev2 * sh1c;
            const float m0b = wD * dot * 0.5773502691896258f;
            atomicAdd(&nsrow[u], m0a);
            atomicAdd(&nsrow[64 + u], m0b);
            const float wBes = wB * es;
            atomicAdd(&nvrow[u * 3 + 0], wBes * sh1a);
            atomicAdd(&nvrow[u * 3 + 1], wBes * sh1b);
            atomicAdd(&nvrow[u * 3 + 2], wBes * sh1c);
            const float wCsh0 = wC * sh0;
            atomicAdd(&nvrow[(64 + u) * 3 + 0], wCsh0 * ev0);
            atomicAdd(&nvrow[(64 + u) * 3 + 1], wCsh0 * ev1);
            atomicAdd(&nvrow[(64 + u) * 3 + 2], wCsh0 * ev2);
        }
    }
}

// ---------------------------------------------------------------------------
// Kernel 3: finalize.  Block = 16 nodes, 4 waves (wave 0: scalar; 1-3: vec d).
// ---------------------------------------------------------------------------
__global__ __launch_bounds__(128) void e3_finalize_kernel(
    const float* __restrict__ ns, const float* __restrict__ nv,
    const float* __restrict__ deg, const float* __restrict__ sc,
    const float* __restrict__ pk,
    const float* __restrict__ g_s, const float* __restrict__ g_v,
    float* __restrict__ out)
{
    __shared__ __align__(16) float amat[4][16 * 132];  // [s, v0, v1, v2]
    __shared__ float dnbuf[16];
    __shared__ float vred[3][16];

    const int tid = threadIdx.x;
    const int n0 = blockIdx.x * 16;

    if (tid < 16) {
        const float dg = deg[n0 + tid];
        dnbuf[tid] = rsqrtf(fmaxf(dg, 1.0f));
    }
    __syncthreads();

    #pragma unroll 2
    for (int idx = tid; idx < 16 * 128; idx += 128) {
        const int r = idx >> 7, c = idx & 127;
        amat[0][r * 132 + c] = ns[(n0 + r) * 128 + c] * dnbuf[r];
    }
    #pragma unroll
    for (int d = 0; d < 3; ++d)
        for (int idx = tid; idx < 16 * 128; idx += 128) {
            const int r = idx >> 7, c = idx & 127;
            amat[1 + d][r * 132 + c] = nv[(n0 + r) * 384 + c * 3 + d] * dnbuf[r];
        }
    __syncthreads();

    const int lane = tid & 31, wave = tid >> 5;
    const int l15 = lane & 15, hi = lane >> 4;
    const float* A = amat[wave];
    const float* Wp = (wave == 0) ? pk + 16384 : pk + 24576;  // packed w2_s / w2_v
    const int d = wave - 1;
    const float inv2 = 0.08838834764831845f;  // 1/sqrt(2*MUL)

    v2f a[32];
    #pragma unroll
    for (int ks = 0; ks < 32; ++ks)
        a[ks] = *(const v2f*)&A[l15 * 132 + ks * 4 + hi * 2];

    v8f res[4];
    #pragma unroll
    for (int nt = 0; nt < 4; ++nt) {
        v8f acc = {};
        const int nn = nt * 16 + l15;
        #pragma unroll
        for (int ks = 0; ks < 32; ++ks)
            acc = wmma_f32(a[ks], loadB(Wp, ks, hi, nn, 64), acc);
        #pragma unroll
        for (int i = 0; i < 8; ++i) {
            const int m = i + hi * 8;
            const float scv = (wave == 0)
                ? sc[(n0 + m) * 256 + nn]
                : sc[(n0 + m) * 256 + 64 + nn * 3 + d];
            acc[i] = acc[i] * inv2 + scv;
        }
        res[nt] = acc;
    }

    float nodeSS[8];
    #pragma unroll
    for (int i = 0; i < 8; ++i) {
        float ss = 0.0f;
        #pragma unroll
        for (int nt = 0; nt < 4; ++nt) { const float v = res[nt][i]; ss += v * v; }
        #pragma unroll
        for (int off = 1; off < 16; off <<= 1)
            ss += __shfl_xor(ss, off, 32);
        nodeSS[i] = ss;
    }

    if (wave > 0 && l15 == 0) {
        #pragma unroll
        for (int i = 0; i < 8; ++i)
            vred[wave - 1][i + hi * 8] = nodeSS[i];
    }
    __syncthreads();

    if (wave == 0) {
        #pragma unroll
        for (int nt = 0; nt < 4; ++nt) {
            const int nn = nt * 16 + l15;
            const float g = g_s[nn];
            #pragma unroll
            for (int i = 0; i < 8; ++i) {
                const int m = i + hi * 8;
                const float rms = rsqrtf(nodeSS[i] * (1.0f / 64.0f) + 1e-5f);
                out[(n0 + m) * 256 + nn] = res[nt][i] * rms * g;
            }
        }
    } else {
        #pragma unroll
        for (int nt = 0; nt < 4; ++nt) {
            const int nn = nt * 16 + l15;
            const float g = g_v[nn];
            #pragma unroll
            for (int i = 0; i < 8; ++i) {
                const int m = i + hi * 8;
                const float tot = vred[0][m] + vred[1][m] + vred[2][m];
                const float rms = rsqrtf(tot * (1.0f / 192.0f) + 1e-5f);
                out[(n0 + m) * 256 + 64 + nn * 3 + d] = res[nt][i] * rms * g;
            }
        }
    }
}

// ---------------------------------------------------------------------------
extern "C" void kernel_launch(void* const* d_in, const int* in_sizes, int n_in,
                              void* d_out, int out_size, void* d_ws, size_t ws_size,
                              hipStream_t stream) {
    const float* nh    = (const float*)d_in[0];
    const float* ea    = (const float*)d_in[1];
    const float* esh   = (const float*)d_in[2];
    const float* w1_s  = (const float*)d_in[3];
    const float* w1_v  = (const float*)d_in[4];
    const float* wsc_s = (const float*)d_in[5];
    const float* wsc_v = (const float*)d_in[6];
    const float* w2_s  = (const float*)d_in[7];
    const float* w2_v  = (const float*)d_in[8];
    const float* fc1   = (const float*)d_in[9];
    const float* fc2   = (const float*)d_in[10];
    const float* fc3   = (const float*)d_in[11];
    const float* g_s   = (const float*)d_in[12];
    const float* g_v   = (const float*)d_in[13];
    const int*   eidx  = (const int*)d_in[14];
    float* out = (float*)d_out;

    float* ws  = (float*)d_ws;
    float* y   = ws;                                   // N x 256
    float* sc  = y   + (size_t)N_NODES * 256;          // N x 256
    float* nsb = sc  + (size_t)N_NODES * 256;          // N x 128
    float* nvb = nsb + (size_t)N_NODES * 128;          // N x 128 x 3
    float* deg = nvb + (size_t)N_NODES * 384;          // N
    float* pkw = deg + (size_t)N_NODES;                // 53760 packed weights

    const int zn = N_NODES * (128 + 384 + 1);          // ns|nv|deg contiguous
    e3_zero_kernel<<<(zn + 255) / 256, 256, 0, stream>>>(nsb, zn);
    e3_pack_kernel<<<(53760 + 255) / 256, 256, 0, stream>>>(
        w1_s, wsc_s, w1_v, wsc_v, w2_s, w2_v, fc1, fc2, fc3, pkw);
    e3_node_gemm_kernel<<<N_NODES / 16, 256, 0, stream>>>(nh, pkw, y, sc);
    e3_edge_kernel<<<N_EDGES / 32, 64, 0, stream>>>(
        ea, esh, eidx, pkw, y, nsb, nvb, deg);
    e3_finalize_kernel<<<N_NODES / 16, 128, 0, stream>>>(
        nsb, nvb, deg, sc, pkw, g_s, g_v, out);
}